// Pointnet2MSG_SUB_90357521973281
// MI455X (gfx1250) — compile-verified
//
#include <hip/hip_runtime.h>
#include <hip/hip_bf16.h>

typedef __attribute__((ext_vector_type(2))) float v2f;
typedef __attribute__((ext_vector_type(8))) float v8f;

// ---------------------------------------------------------------------------
// Farthest point sampling: one workgroup per batch, dists live in LDS.
// ---------------------------------------------------------------------------
__global__ void __launch_bounds__(1024) fps_kernel(const float* __restrict__ pts,
                                                   int pstride, int n, int npoint,
                                                   int* __restrict__ out) {
  __shared__ float sdist[16384];
  __shared__ float rval[1024];
  __shared__ int   ridx[1024];
  __shared__ float spt[3];
  const int b = blockIdx.x;
  const float* Pb = pts + (size_t)b * n * pstride;
  int* Ob = out + (size_t)b * npoint;
  const int tid = threadIdx.x;
  for (int i = tid; i < n; i += blockDim.x) sdist[i] = 1e10f;
  if (tid == 0) { Ob[0] = 0; spt[0] = Pb[0]; spt[1] = Pb[1]; spt[2] = Pb[2]; }
  __syncthreads();
  for (int it = 1; it < npoint; ++it) {
    const float lx = spt[0], ly = spt[1], lz = spt[2];
    float bestv = -1.f; int besti = 0;
    for (int i = tid; i < n; i += blockDim.x) {
      const float* q = Pb + (size_t)i * pstride;
      float dx = q[0] - lx, dy = q[1] - ly, dz = q[2] - lz;
      float d = dx * dx + dy * dy + dz * dz;
      float old = sdist[i];
      d = d < old ? d : old;
      sdist[i] = d;
      if (d > bestv) { bestv = d; besti = i; }
    }
    rval[tid] = bestv; ridx[tid] = besti;
    __syncthreads();
    for (int s = blockDim.x >> 1; s > 0; s >>= 1) {
      if (tid < s) {
        float v2 = rval[tid + s]; int i2 = ridx[tid + s];
        if (v2 > rval[tid] || (v2 == rval[tid] && i2 < ridx[tid])) {
          rval[tid] = v2; ridx[tid] = i2;
        }
      }
      __syncthreads();
    }
    if (tid == 0) {
      int c = ridx[0]; Ob[it] = c;
      const float* q = Pb + (size_t)c * pstride;
      spt[0] = q[0]; spt[1] = q[1]; spt[2] = q[2];
    }
    __syncthreads();
  }
}

// out[b,m,:] = pts[b, idx[b,m], 0..2]
__global__ void gather_xyz_kernel(const float* __restrict__ pts, int pstride, int n,
                                  const int* __restrict__ idx, int M,
                                  float* __restrict__ out) {
  const int b = blockIdx.y;
  int t = blockIdx.x * blockDim.x + threadIdx.x;
  if (t >= M * 3) return;
  int m = t / 3, c = t % 3;
  int i = idx[(size_t)b * M + m];
  out[((size_t)b * M + m) * 3 + c] = pts[((size_t)b * n + i) * pstride + c];
}

// feat0[b,c,n] = pc[b,n,3+c]
__global__ void extract_feat0_kernel(const float* __restrict__ pc, int N,
                                     float* __restrict__ feat0) {
  const int b = blockIdx.y;
  int t = blockIdx.x * blockDim.x + threadIdx.x;
  if (t >= 3 * N) return;
  int c = t / N, i = t % N;
  feat0[((size_t)b * 3 + c) * N + i] = pc[((size_t)b * N + i) * 6 + 3 + c];
}

// out[b,i,c] = pc[b,i,c] for c<3
__global__ void copy_xyz_out_kernel(const float* __restrict__ pc, int N,
                                    float* __restrict__ out) {
  const int b = blockIdx.y;
  int t = blockIdx.x * blockDim.x + threadIdx.x;
  if (t >= N * 3) return;
  out[(size_t)b * N * 3 + t] = pc[(size_t)b * N * 6 + (size_t)(t / 3) * 6 + (t % 3)];
}

// ---------------------------------------------------------------------------
// Ball query: one wave per center; first `ns` in-radius points in index order.
// ---------------------------------------------------------------------------
__global__ void ball_query_kernel(const float* __restrict__ pxyz, int pstride, int n,
                                  const float* __restrict__ q, int M, float r2,
                                  int ns, int* __restrict__ out) {
  int wid = (blockIdx.x * blockDim.x + threadIdx.x) >> 5;
  int lane = threadIdx.x & 31;
  int b = wid / M, m = wid % M;
  float cx = q[((size_t)b * M + m) * 3 + 0];
  float cy = q[((size_t)b * M + m) * 3 + 1];
  float cz = q[((size_t)b * M + m) * 3 + 2];
  const float* Pb = pxyz + (size_t)b * n * pstride;
  int* Ob = out + ((size_t)b * M + m) * ns;
  int cnt = 0, firstidx = -1;
  for (int base = 0; base < n; base += 32) {
    int i = base + lane;
    bool in = false;
    if (i < n) {
      const float* p = Pb + (size_t)i * pstride;
      float dx = p[0] - cx, dy = p[1] - cy, dz = p[2] - cz;
      in = (dx * dx + dy * dy + dz * dz) < r2;
    }
    unsigned mask = (unsigned)__ballot(in);
    if (firstidx < 0 && mask) firstidx = base + __ffs(mask) - 1;
    if (in) {
      int pos = cnt + __popc(mask & ((1u << lane) - 1u));
      if (pos < ns) Ob[pos] = i;
    }
    cnt += __popc(mask);
    if (cnt >= ns) break;
  }
  if (cnt > ns) cnt = ns;
  if (firstidx < 0) firstidx = 0;
  for (int pos = cnt + lane; pos < ns; pos += 32) Ob[pos] = firstidx;
}

// ---------------------------------------------------------------------------
// Grouping: X[b, c, m*ns+s]; rows 0..2 centered xyz, 3..3+Cfeat-1 features,
// remaining (K-pad) rows = 0.
// ---------------------------------------------------------------------------
__global__ void group_kernel(const float* __restrict__ pxyz, int pstride, int n,
                             const float* __restrict__ feat, int featLD, int Cfeat,
                             const float* __restrict__ centers,
                             const int* __restrict__ idx, int M, int ns, int Cpad,
                             float* __restrict__ out) {
  const int b = blockIdx.y;
  const int P = M * ns;
  int t = blockIdx.x * blockDim.x + threadIdx.x;
  if (t >= Cpad * P) return;
  int c = t / P, p = t % P;
  int m = p / ns, s = p % ns;
  int i = idx[((size_t)b * M + m) * ns + s];
  float v = 0.f;
  if (c < 3)
    v = pxyz[((size_t)b * n + i) * pstride + c] - centers[((size_t)b * M + m) * 3 + c];
  else if (c < 3 + Cfeat)
    v = feat[((size_t)b * Cfeat + (c - 3)) * featLD + i];
  out[((size_t)b * Cpad + c) * (size_t)P + p] = v;
}

// Zero-pad weights into [O, Cpad] so the GEMM K-loop needs no bounds guards.
__global__ void pack_weights_kernel(const float* __restrict__ W, int O, int Cin,
                                    int Cpad, float* __restrict__ Wpk) {
  int t = blockIdx.x * blockDim.x + threadIdx.x;
  if (t >= O * Cpad) return;
  int o = t / Cpad, c = t % Cpad;
  Wpk[t] = (c < Cin) ? W[(size_t)o * Cin + c] : 0.f;
}

// ---------------------------------------------------------------------------
// GEMM + bias + ReLU on V_WMMA_F32_16X16X4_F32.
// Y[b,o,p] = relu(bias[o] + sum_c Wpk[o,c] * X[b,c,p]).
// 4 waves/block, one 16x16 tile per wave along P. Crows (= padded K, zero-
// filled in both Wpk and X) is a multiple of 4; main loop does K-step 8 with
// two WMMAs, tail handles Crows%8==4. No guards -> EXEC stays all-ones.
// A-operand pair is contiguous in packed weights -> single b64 load.
// ---------------------------------------------------------------------------
__global__ void __launch_bounds__(128) gemm_bias_relu_wmma(
    const float* __restrict__ X, const float* __restrict__ Wpk,
    const float* __restrict__ bias, float* __restrict__ Y,
    int Crows, int P, int O) {
  const int b = blockIdx.z;
  const int wave = threadIdx.x >> 5;
  const int p0 = (blockIdx.x * 4 + wave) << 4;
  const int o0 = blockIdx.y << 4;
  const int lane = threadIdx.x & 31;
  const int ln = lane & 15;
  const int half = lane >> 4;
  const size_t Ps = (size_t)P;
  const float* aptr = Wpk + (size_t)(o0 + ln) * Crows + (half << 1);
  const float* xptr = X + (size_t)b * Crows * Ps + (size_t)(half << 1) * Ps + p0 + ln;
  const size_t xstep4 = 4 * Ps;
  v8f acc = {};
  int c = 0;
  for (; c + 8 <= Crows; c += 8) {
    v2f a0 = *(const v2f*)(aptr);          // W[o, c+2h .. c+2h+1]
    v2f a1 = *(const v2f*)(aptr + 4);      // W[o, c+4+2h .. c+4+2h+1]
    v2f b0, b1;
    b0.x = xptr[0];                        // X[c+2h,   p]
    b0.y = xptr[Ps];                       // X[c+2h+1, p]
    b1.x = xptr[xstep4];
    b1.y = xptr[xstep4 + Ps];
    acc = __builtin_amdgcn_wmma_f32_16x16x4_f32(false, a0, false, b0,
                                                (short)0, acc, false, false);
    acc = __builtin_amdgcn_wmma_f32_16x16x4_f32(false, a1, false, b1,
                                                (short)0, acc, false, false);
    aptr += 8;
    xptr += 2 * xstep4;
  }
  if (c < Crows) {                         // single K=4 tail (Crows % 8 == 4)
    v2f a0 = *(const v2f*)(aptr);
    v2f b0;
    b0.x = xptr[0];
    b0.y = xptr[Ps];
    acc = __builtin_amdgcn_wmma_f32_16x16x4_f32(false, a0, false, b0,
                                                (short)0, acc, false, false);
  }
  float* Yb = Y + (size_t)b * O * Ps;
  #pragma unroll
  for (int r = 0; r < 8; ++r) {
    int o = o0 + r + (half << 3);
    float v = acc[r] + bias[o];
    Yb[(size_t)o * Ps + p0 + ln] = v > 0.f ? v : 0.f;
  }
}

// dst[b, co+o, m] = max_s x[b, o, m*ns+s]
__global__ void maxpool_kernel(const float* __restrict__ x, int O, int M, int ns,
                               float* __restrict__ dst, int co, int Ctot) {
  const int b = blockIdx.y;
  int t = blockIdx.x * blockDim.x + threadIdx.x;
  if (t >= O * M) return;
  int o = t / M, m = t % M;
  const float* xr = x + ((size_t)b * O + o) * (size_t)(M * ns) + (size_t)m * ns;
  float v = xr[0];
  for (int s = 1; s < ns; ++s) v = fmaxf(v, xr[s]);
  dst[((size_t)b * Ctot + co + o) * M + m] = v;
}

// 3-NN (squared distances) + normalized inverse-distance weights.
__global__ void three_nn_kernel(const float* __restrict__ dxyz,
                                const float* __restrict__ sxyz, int Nd, int Ns,
                                int* __restrict__ oidx, float* __restrict__ ow) {
  const int b = blockIdx.y;
  int i = blockIdx.x * blockDim.x + threadIdx.x;
  if (i >= Nd) return;
  const float* dp = dxyz + ((size_t)b * Nd + i) * 3;
  float x = dp[0], y = dp[1], z = dp[2];
  float d0 = 3e38f, d1 = 3e38f, d2 = 3e38f;
  int i0 = 0, i1 = 0, i2 = 0;
  const float* sp = sxyz + (size_t)b * Ns * 3;
  for (int j = 0; j < Ns; ++j) {
    float dx = x - sp[3 * j], dy = y - sp[3 * j + 1], dz = z - sp[3 * j + 2];
    float d = dx * dx + dy * dy + dz * dz;
    if (d < d0)      { d2 = d1; i2 = i1; d1 = d0; i1 = i0; d0 = d; i0 = j; }
    else if (d < d1) { d2 = d1; i2 = i1; d1 = d;  i1 = j; }
    else if (d < d2) { d2 = d;  i2 = j; }
  }
  float r0 = 1.f / (d0 + 1e-8f), r1 = 1.f / (d1 + 1e-8f), r2 = 1.f / (d2 + 1e-8f);
  float s = r0 + r1 + r2;
  size_t o = ((size_t)b * Nd + i) * 3;
  oidx[o] = i0; oidx[o + 1] = i1; oidx[o + 2] = i2;
  ow[o] = r0 / s; ow[o + 1] = r1 / s; ow[o + 2] = r2 / s;
}

// dst rows [0..C) of concat buffer = interpolated features.
__global__ void interp_kernel(const float* __restrict__ feat2, int C, int Ns,
                              const int* __restrict__ oidx,
                              const float* __restrict__ ow, int Nd,
                              float* __restrict__ dst, int CpadRows) {
  const int b = blockIdx.y;
  int t = blockIdx.x * blockDim.x + threadIdx.x;
  if (t >= C * Nd) return;
  int c = t / Nd, i = t % Nd;
  size_t ko = ((size_t)b * Nd + i) * 3;
  const float* f = feat2 + ((size_t)b * C + c) * Ns;
  float v = f[oidx[ko]] * ow[ko] + f[oidx[ko + 1]] * ow[ko + 1] +
            f[oidx[ko + 2]] * ow[ko + 2];
  dst[((size_t)b * CpadRows + c) * Nd + i] = v;
}

// dst rows [rowOff..rowOff+C) = src rows [0..C)
__global__ void copy_rows_kernel(const float* __restrict__ src, int C, int LD,
                                 float* __restrict__ dst, int rowOff, int CpadRows) {
  const int b = blockIdx.y;
  int t = blockIdx.x * blockDim.x + threadIdx.x;
  if (t >= C * LD) return;
  int c = t / LD, i = t % LD;
  dst[((size_t)b * CpadRows + rowOff + c) * LD + i] =
      src[((size_t)b * C + c) * LD + i];
}

__global__ void zero_rows_kernel(float* __restrict__ dst, int CpadRows, int LD,
                                 int row0, int nrows) {
  const int b = blockIdx.y;
  int t = blockIdx.x * blockDim.x + threadIdx.x;
  if (t >= nrows * LD) return;
  int c = t / LD, i = t % LD;
  dst[((size_t)b * CpadRows + row0 + c) * LD + i] = 0.f;
}

// ---------------------------------------------------------------------------
extern "C" void kernel_launch(void* const* d_in, const int* in_sizes, int n_in,
                              void* d_out, int out_size, void* d_ws, size_t ws_size,
                              hipStream_t stream) {
  const int Bb = 2, N = 16384, M1 = 1024, M2 = 256;

  // ---- locate pointcloud slot by its unique flat size --------------------
  int pcIdx = 0;
  for (int i = 0; i < n_in; ++i)
    if (in_sizes[i] == Bb * N * 6) { pcIdx = i; break; }
  const float* pc = (const float*)d_in[pcIdx];

  // ---- recover weight/bias pointers (order + w/b auto-detected) ----------
  struct Dim { int o, c; };
  // canonical: sa[0][0]L0..L2, sa[0][1], sa[1][0], sa[1][1], fp[0]L0..1, fp[1]L0..1
  Dim dims[16] = {{16, 6},  {16, 16}, {32, 16},  {32, 6},  {32, 32}, {64, 32},
                  {64, 99}, {64, 64}, {128, 64}, {64, 99}, {96, 64}, {128, 96},
                  {128, 259}, {128, 128}, {256, 352}, {256, 256}};
  int cpad[16] = {8, 16, 16, 8, 32, 32, 100, 64, 64, 100, 64, 96,
                  260, 128, 352, 256};
  const float* Wp[16]; const float* Bp[16];
  int slots[64]; int nslots = 0;
  for (int i = 0; i < n_in && nslots < 64; ++i)
    if (i != pcIdx) slots[nslots++] = i;
  auto pairMatches = [&](int s, Dim d) {
    if (s + 1 >= nslots) return false;
    int a = in_sizes[slots[s]], b = in_sizes[slots[s + 1]];
    return (a == d.o * d.c && b == d.o) || (a == d.o && b == d.o * d.c);
  };
  bool saFirst = pairMatches(0, dims[0]);
  int order[16];
  if (saFirst) { for (int i = 0; i < 16; ++i) order[i] = i; }
  else { // fp params flattened first
    order[0] = 12; order[1] = 13; order[2] = 14; order[3] = 15;
    for (int i = 0; i < 12; ++i) order[4 + i] = i;
  }
  int cursor = 0;
  for (int k = 0; k < 16 && cursor + 1 < nslots; ++k) {
    int ci = order[k];
    int a = slots[cursor], b = slots[cursor + 1];
    cursor += 2;
    if (in_sizes[a] == dims[ci].o * dims[ci].c) {
      Wp[ci] = (const float*)d_in[a]; Bp[ci] = (const float*)d_in[b];
    } else {
      Wp[ci] = (const float*)d_in[b]; Bp[ci] = (const float*)d_in[a];
    }
  }

  // ---- workspace layout ---------------------------------------------------
  size_t off = 0;
  auto wsAlloc = [&](size_t bytes) -> void* {
    void* p = (char*)d_ws + off;
    off += (bytes + 255) & ~(size_t)255;
    return p;
  };
  int*   fidx1  = (int*)  wsAlloc(sizeof(int)   * Bb * M1);
  int*   fidx2  = (int*)  wsAlloc(sizeof(int)   * Bb * M2);
  float* nxyz1  = (float*)wsAlloc(sizeof(float) * Bb * M1 * 3);
  float* nxyz2  = (float*)wsAlloc(sizeof(float) * Bb * M2 * 3);
  float* feat0  = (float*)wsAlloc(sizeof(float) * Bb * 3 * N);
  int*   bq00   = (int*)  wsAlloc(sizeof(int)   * Bb * M1 * 16);
  int*   bq01   = (int*)  wsAlloc(sizeof(int)   * Bb * M1 * 32);
  int*   bq10   = (int*)  wsAlloc(sizeof(int)   * Bb * M2 * 16);
  int*   bq11   = (int*)  wsAlloc(sizeof(int)   * Bb * M2 * 32);
  float* feat1  = (float*)wsAlloc(sizeof(float) * Bb * 96  * M1);
  float* feat2  = (float*)wsAlloc(sizeof(float) * Bb * 256 * M2);
  float* feat1b = (float*)wsAlloc(sizeof(float) * Bb * 256 * M1);
  float* xg     = (float*)wsAlloc(sizeof(float) * Bb * 100 * 8192);   // max Cpad*P
  float* actA   = (float*)wsAlloc(sizeof(float) * (size_t)Bb * 64 * 32768);
  float* actB   = (float*)wsAlloc(sizeof(float) * (size_t)Bb * 64 * 32768);
  float* ccat   = (float*)wsAlloc(sizeof(float) * (size_t)Bb * 260 * N);
  int*   knnI   = (int*)  wsAlloc(sizeof(int)   * Bb * N * 3);
  float* knnW   = (float*)wsAlloc(sizeof(float) * Bb * N * 3);
  (void)ws_size; (void)out_size;

  // ---- pack (zero-pad) all layer weights once ----------------------------
  float* Wpk[16];
  for (int i = 0; i < 16; ++i) {
    int sz = dims[i].o * cpad[i];
    Wpk[i] = (float*)wsAlloc(sizeof(float) * sz);
    pack_weights_kernel<<<(sz + 255) / 256, 256, 0, stream>>>(
        Wp[i], dims[i].o, dims[i].c, cpad[i], Wpk[i]);
  }

  float* outXyz  = (float*)d_out;
  float* outFeat = (float*)d_out + (size_t)Bb * N * 3;

  auto blk = [](int total) { return (total + 255) / 256; };
  auto gemm = [&](const float* X, int li, float* Y, int P, int O) {
    dim3 g(P / 64, O / 16, Bb);
    gemm_bias_relu_wmma<<<g, dim3(128), 0, stream>>>(X, Wpk[li], Bp[li], Y,
                                                     cpad[li], P, O);
  };

  // ---- stage 0: xyz/feature extraction -----------------------------------
  copy_xyz_out_kernel<<<dim3(blk(N * 3), Bb), 256, 0, stream>>>(pc, N, outXyz);
  extract_feat0_kernel<<<dim3(blk(3 * N), Bb), 256, 0, stream>>>(pc, N, feat0);

  // ---- FPS levels ---------------------------------------------------------
  fps_kernel<<<Bb, 1024, 0, stream>>>(pc, 6, N, M1, fidx1);
  gather_xyz_kernel<<<dim3(blk(M1 * 3), Bb), 256, 0, stream>>>(pc, 6, N, fidx1, M1, nxyz1);
  fps_kernel<<<Bb, 1024, 0, stream>>>(nxyz1, 3, M1, M2, fidx2);
  gather_xyz_kernel<<<dim3(blk(M2 * 3), Bb), 256, 0, stream>>>(nxyz1, 3, M1, fidx2, M2, nxyz2);

  // ---- SA1 (N=16384 -> M1=1024, radii 0.5/1.0, ns 16/32) ------------------
  ball_query_kernel<<<Bb * M1 / 4, 128, 0, stream>>>(pc, 6, N, nxyz1, M1, 0.25f, 16, bq00);
  ball_query_kernel<<<Bb * M1 / 4, 128, 0, stream>>>(pc, 6, N, nxyz1, M1, 1.00f, 32, bq01);

  { // branch 0: [6,16,16,32], P = 1024*16
    const int P = M1 * 16, Cp = 8;
    group_kernel<<<dim3(blk(Cp * P), Bb), 256, 0, stream>>>(pc, 6, N, feat0, N, 3,
                                                            nxyz1, bq00, M1, 16, Cp, xg);
    gemm(xg, 0, actA, P, 16);
    gemm(actA, 1, actB, P, 16);
    gemm(actB, 2, actA, P, 32);
    maxpool_kernel<<<dim3(blk(32 * M1), Bb), 256, 0, stream>>>(actA, 32, M1, 16, feat1, 0, 96);
  }
  { // branch 1: [6,32,32,64], P = 1024*32
    const int P = M1 * 32, Cp = 8;
    group_kernel<<<dim3(blk(Cp * P), Bb), 256, 0, stream>>>(pc, 6, N, feat0, N, 3,
                                                            nxyz1, bq01, M1, 32, Cp, xg);
    gemm(xg, 3, actA, P, 32);
    gemm(actA, 4, actB, P, 32);
    gemm(actB, 5, actA, P, 64);
    maxpool_kernel<<<dim3(blk(64 * M1), Bb), 256, 0, stream>>>(actA, 64, M1, 32, feat1, 32, 96);
  }

  // ---- SA2 (M1=1024 -> M2=256, radii 1.0/2.0, ns 16/32) -------------------
  ball_query_kernel<<<Bb * M2 / 4, 128, 0, stream>>>(nxyz1, 3, M1, nxyz2, M2, 1.0f, 16, bq10);
  ball_query_kernel<<<Bb * M2 / 4, 128, 0, stream>>>(nxyz1, 3, M1, nxyz2, M2, 4.0f, 32, bq11);

  { // branch 0: [99,64,64,128], P = 256*16
    const int P = M2 * 16, Cp = 100;
    group_kernel<<<dim3(blk(Cp * P), Bb), 256, 0, stream>>>(nxyz1, 3, M1, feat1, M1, 96,
                                                            nxyz2, bq10, M2, 16, Cp, xg);
    gemm(xg, 6, actA, P, 64);
    gemm(actA, 7, actB, P, 64);
    gemm(actB, 8, actA, P, 128);
    maxpool_kernel<<<dim3(blk(128 * M2), Bb), 256, 0, stream>>>(actA, 128, M2, 16, feat2, 0, 256);
  }
  { // branch 1: [99,64,96,128], P = 256*32
    const int P = M2 * 32, Cp = 100;
    group_kernel<<<dim3(blk(Cp * P), Bb), 256, 0, stream>>>(nxyz1, 3, M1, feat1, M1, 96,
                                                            nxyz2, bq11, M2, 32, Cp, xg);
    gemm(xg, 9, actA, P, 64);
    gemm(actA, 10, actB, P, 96);
    gemm(actB, 11, actA, P, 128);
    maxpool_kernel<<<dim3(blk(128 * M2), Bb), 256, 0, stream>>>(actA, 128, M2, 32, feat2, 128, 256);
  }

  // ---- FP stage 1: (nxyz1, nxyz2, feat1, feat2) -> feat1b [B,256,1024] ----
  {
    const int Nd = M1, Cp = 352;
    three_nn_kernel<<<dim3(blk(Nd), Bb), 256, 0, stream>>>(nxyz1, nxyz2, Nd, M2, knnI, knnW);
    interp_kernel<<<dim3(blk(256 * Nd), Bb), 256, 0, stream>>>(feat2, 256, M2, knnI, knnW,
                                                               Nd, ccat, Cp);
    copy_rows_kernel<<<dim3(blk(96 * Nd), Bb), 256, 0, stream>>>(feat1, 96, Nd, ccat, 256, Cp);
    gemm(ccat, 14, actA, Nd, 256);
    gemm(actA, 15, feat1b, Nd, 256);
  }

  // ---- FP stage 2: (xyz, nxyz1, feat0, feat1b) -> d_out [B,128,16384] -----
  {
    const int Nd = N, Cp = 260;
    three_nn_kernel<<<dim3(blk(Nd), Bb), 256, 0, stream>>>(outXyz, nxyz1, Nd, M1, knnI, knnW);
    interp_kernel<<<dim3(blk(256 * Nd), Bb), 256, 0, stream>>>(feat1b, 256, M1, knnI, knnW,
                                                               Nd, ccat, Cp);
    copy_rows_kernel<<<dim3(blk(3 * Nd), Bb), 256, 0, stream>>>(feat0, 3, Nd, ccat, 256, Cp);
    zero_rows_kernel<<<dim3(blk(1 * Nd), Bb), 256, 0, stream>>>(ccat, Cp, Nd, 259, 1);
    gemm(ccat, 12, actA, Nd, 128);
    gemm(actA, 13, outFeat, Nd, 128);
  }
}